// AttentionLayer_78786880078387
// MI455X (gfx1250) — compile-verified
//
#include <hip/hip_runtime.h>
#include <math.h>

typedef float v2f __attribute__((ext_vector_type(2)));
typedef float v8f __attribute__((ext_vector_type(8)));

#define B_  8
#define TE_ 256
#define TD_ 256
#define HE_ 512
#define HD_ 512

// ---------------------------------------------------------------------------
// Generic f32 WMMA GEMM:  C[M,N] = A[M,K] * B[K,N]   (row-major, tight pitch)
// 32x32 tile per block, 4 waves (128 threads), each wave a 16x16 sub-tile,
// K stepped 32 at a time through LDS, inner loop of v_wmma_f32_16x16x4_f32.
// blockIdx.z selects batch via explicit strides (0 strides => single GEMM).
// ---------------------------------------------------------------------------
__global__ __launch_bounds__(128)
void wmma_gemm_f32(const float* __restrict__ A, const float* __restrict__ Bm,
                   float* __restrict__ C, int M, int N, int K,
                   long long aBatch, long long bBatch, long long cBatch)
{
    A  += (long long)blockIdx.z * aBatch;
    Bm += (long long)blockIdx.z * bBatch;
    C  += (long long)blockIdx.z * cBatch;

    // pitch 34 floats: keeps ds_load_b64 8B-aligned per row and bank-conflict free
    __shared__ __align__(16) float As[32][34];
    __shared__ __align__(16) float Bs[32][34];   // transposed: Bs[col][k]

    const int tid  = threadIdx.x;
    const int wid  = tid >> 5;          // wave 0..3
    const int lane = tid & 31;
    const int half = lane >> 4;         // 0: K lo pair / rows 0-7, 1: K hi pair / rows 8-15
    const int lh   = lane & 15;
    const int wr   = (wid >> 1) * 16;   // wave row offset inside 32x32 tile
    const int wc   = (wid & 1)  * 16;   // wave col offset

    const int m0 = blockIdx.y * 32;
    const int n0 = blockIdx.x * 32;

    v8f acc = {};

    for (int k0 = 0; k0 < K; k0 += 32) {
        // stage A tile (32 rows x 32 k) : 256 float4, 2 per thread
        #pragma unroll
        for (int p = 0; p < 2; ++p) {
            int idx = tid * 2 + p;                 // 0..255
            int r   = idx >> 3;                    // 0..31
            int kq  = (idx & 7) << 2;              // 0,4,...,28
            const float4 va = *(const float4*)(&A[(long long)(m0 + r) * K + k0 + kq]);
            As[r][kq + 0] = va.x; As[r][kq + 1] = va.y;
            As[r][kq + 2] = va.z; As[r][kq + 3] = va.w;
        }
        // stage B tile transposed: B[k0+r][n0+c] -> Bs[c][r]
        #pragma unroll
        for (int p = 0; p < 2; ++p) {
            int idx = tid * 2 + p;
            int r   = idx >> 3;                    // k 0..31
            int cq  = (idx & 7) << 2;              // col 0,4,...,28
            const float4 vb = *(const float4*)(&Bm[(long long)(k0 + r) * N + n0 + cq]);
            Bs[cq + 0][r] = vb.x; Bs[cq + 1][r] = vb.y;
            Bs[cq + 2][r] = vb.z; Bs[cq + 3][r] = vb.w;
        }
        __syncthreads();

        // 8 WMMA steps of K=4 each.
        // A frag (ISA 16x4 f32 layout): v0 = A[m=lh][k+2*half], v1 = A[m=lh][k+2*half+1]
        // B frag mirrors it from the transposed LDS tile.
        #pragma unroll
        for (int kk = 0; kk < 32; kk += 4) {
            v2f a = *(const v2f*)(&As[wr + lh][kk + 2 * half]);
            v2f b = *(const v2f*)(&Bs[wc + lh][kk + 2 * half]);
            acc = __builtin_amdgcn_wmma_f32_16x16x4_f32(
                      false, a, false, b, (short)0, acc, false, false);
        }
        __syncthreads();
    }

    // C/D 16x16 f32 layout: vgpr i -> row (i + 8*half), col lh
    #pragma unroll
    for (int i = 0; i < 8; ++i) {
        C[(long long)(m0 + wr + i + half * 8) * N + (n0 + wc + lh)] = acc[i];
    }
}

// ---------------------------------------------------------------------------
// Fused scores + softmax.
// One block per (b, d): thread t computes
//   s(t) = sum_k tanh(Ws[b,t,k] + Uh[b,d,k]) * V[k]
// then block-wide softmax over t (TE=256), writing energies e[b,d,:].
// Ws is streamed through LDS in coalesced K-chunks; Uh row and V broadcast
// from LDS (same-address -> LDS broadcast, conflict free).
// ---------------------------------------------------------------------------
#define KC 16
__global__ __launch_bounds__(256)
void scores_softmax(const float* __restrict__ Ws,   // [B, TE, HE]
                    const float* __restrict__ Uh,   // [B, TD, HE]
                    const float* __restrict__ Va,   // [HE]
                    float* __restrict__ Eout)       // [B, TD, TE]
{
    const int d = blockIdx.x;
    const int b = blockIdx.y;
    const int t = threadIdx.x;

    __shared__ float uh_s[HE_];
    __shared__ float v_s[HE_];
    __shared__ __align__(16) float ws_s[TE_][KC + 1];  // pitch 17: conflict-free column reads
    __shared__ float red[TE_];

    const float* uhrow = Uh + ((long long)b * TD_ + d) * HE_;
    for (int i = t; i < HE_; i += 256) { uh_s[i] = uhrow[i]; v_s[i] = Va[i]; }

    float s = 0.f;
    const float* wsb = Ws + (long long)b * TE_ * HE_;

    for (int k0 = 0; k0 < HE_; k0 += KC) {
        // stage Ws chunk [256 t][16 k] : 1024 float4-lanes worth -> 4 passes
        #pragma unroll
        for (int p = 0; p < 4; ++p) {
            int idx = p * 256 + t;          // 0..1023
            int rr  = idx >> 2;             // t row 0..255
            int kq  = (idx & 3) << 2;       // 0,4,8,12
            const float4 v4 = *(const float4*)(&wsb[(long long)rr * HE_ + k0 + kq]);
            ws_s[rr][kq + 0] = v4.x; ws_s[rr][kq + 1] = v4.y;
            ws_s[rr][kq + 2] = v4.z; ws_s[rr][kq + 3] = v4.w;
        }
        __syncthreads();
        #pragma unroll
        for (int kk = 0; kk < KC; ++kk) {
            s += tanhf(ws_s[t][kk] + uh_s[k0 + kk]) * v_s[k0 + kk];
        }
        __syncthreads();
    }

    // ---- softmax over the 256 scores held one-per-thread ----
    red[t] = s;
    __syncthreads();
    for (int off = 128; off > 0; off >>= 1) {
        if (t < off) red[t] = fmaxf(red[t], red[t + off]);
        __syncthreads();
    }
    const float mx = red[0];
    __syncthreads();

    const float ev = __expf(s - mx);
    red[t] = ev;
    __syncthreads();
    for (int off = 128; off > 0; off >>= 1) {
        if (t < off) red[t] += red[t + off];
        __syncthreads();
    }
    const float denom = red[0];

    Eout[((long long)b * TD_ + d) * TE_ + t] = ev / denom;
}

// ---------------------------------------------------------------------------
extern "C" void kernel_launch(void* const* d_in, const int* in_sizes, int n_in,
                              void* d_out, int out_size, void* d_ws, size_t ws_size,
                              hipStream_t stream) {
    (void)in_sizes; (void)n_in; (void)out_size; (void)ws_size;

    const float* enc = (const float*)d_in[0];   // [8,256,512]
    const float* dec = (const float*)d_in[1];   // [8,256,512]
    const float* W_a = (const float*)d_in[2];   // [512,512]
    const float* U_a = (const float*)d_in[3];   // [512,512]
    const float* V_a = (const float*)d_in[4];   // [512,1]

    float* out   = (float*)d_out;
    float* c_out = out;                                   // [8,256,512]
    float* e_out = out + (long long)B_ * TD_ * HE_;       // [8,256,256]

    float* Ws = (float*)d_ws;                             // [8,256,512] = 4 MB
    float* Uh = Ws + (long long)B_ * TE_ * HE_;           // [8,256,512] = 4 MB

    const dim3 gblk(128);

    // Ws = enc @ W_a : M = B*TE = 2048, N = 512, K = 512
    wmma_gemm_f32<<<dim3(HE_ / 32, (B_ * TE_) / 32, 1), gblk, 0, stream>>>(
        enc, W_a, Ws, B_ * TE_, HE_, HE_, 0, 0, 0);

    // Uh = dec @ U_a : M = B*TD = 2048, N = 512, K = 512
    wmma_gemm_f32<<<dim3(HE_ / 32, (B_ * TD_) / 32, 1), gblk, 0, stream>>>(
        dec, U_a, Uh, B_ * TD_, HE_, HD_, 0, 0, 0);

    // e = softmax_t( sum_k tanh(Ws + Uh) * V )
    scores_softmax<<<dim3(TD_, B_), dim3(256), 0, stream>>>(Ws, Uh, V_a, e_out);

    // c[b] = e[b] @ enc[b] : M = TD = 256, N = HE = 512, K = TE = 256, batched over B
    wmma_gemm_f32<<<dim3(HE_ / 32, TD_ / 32, B_), gblk, 0, stream>>>(
        e_out, enc, c_out, TD_, HE_, TE_,
        (long long)TD_ * TE_, (long long)TE_ * HE_, (long long)TD_ * HE_);
}